// GCN_89043261980674
// MI455X (gfx1250) — compile-verified
//
#include <hip/hip_runtime.h>
#include <math.h>

#define N_NODES 100000
#define N_EDGES 640000
#define F_IN 128
#define F_HID 128
#define F_OUT 40
#define F_OUT_PAD 48
#define BN_EPS 1e-5f

typedef float v2f __attribute__((ext_vector_type(2)));
typedef float v8f __attribute__((ext_vector_type(8)));

// ---------------- degree / dinv ----------------
__global__ void deg_init_kernel(float* deg, int n) {
    int i = blockIdx.x * blockDim.x + threadIdx.x;
    if (i < n) deg[i] = 1.0f;  // self loop
}

__global__ void deg_accum_kernel(float* deg, const int* __restrict__ dst, int e) {
    int i = blockIdx.x * blockDim.x + threadIdx.x;
    if (i < e) unsafeAtomicAdd(&deg[dst[i]], 1.0f);
}

__global__ void dinv_kernel(float* deg, int n) {
    int i = blockIdx.x * blockDim.x + threadIdx.x;
    if (i < n) deg[i] = rsqrtf(fmaxf(deg[i], 1.0f));  // in place: deg -> dinv
}

// ---------------- pad W3 (128x40 -> 128x48, zeros) ----------------
__global__ void pad_w3_kernel(const float* __restrict__ w3, float* __restrict__ w3p) {
    int i = blockIdx.x * blockDim.x + threadIdx.x;
    if (i >= F_HID * F_OUT_PAD) return;
    int k = i / F_OUT_PAD, n = i % F_OUT_PAD;
    w3p[i] = (n < F_OUT) ? w3[k * F_OUT + n] : 0.0f;
}

// ---------------- WMMA fp32 GEMM: H[rb*16.., col0..] = (X @ W) * dinv[row] ----
// X: nRows x 128 (row-major), W: 128 x ldw, H: nRows x ldw
// One wave computes a 16 x (NT*16) output strip. 8 waves per block.
template <int NT>
__global__ void gemm_dinv_kernel(const float* __restrict__ X,
                                 const float* __restrict__ Wm,
                                 const float* __restrict__ dinv,
                                 float* __restrict__ H,
                                 int ldw, int nRowBlocks) {
    int wave = threadIdx.x >> 5;
    int lane = threadIdx.x & 31;
    int rb = blockIdx.x * 8 + wave;
    if (rb >= nRowBlocks) return;  // wave-uniform: EXEC stays all-1s for WMMA
    int col0 = blockIdx.y * (NT * 16);

    int m  = lane & 15;   // A row / B col within tile
    int hi = lane >> 4;   // K-pair selector

    const float2* xr2 = (const float2*)(X + (size_t)(rb * 16 + m) * F_IN);

    v8f acc[NT];
#pragma unroll
    for (int j = 0; j < NT; ++j) acc[j] = (v8f){0, 0, 0, 0, 0, 0, 0, 0};

    for (int k = 0; k < F_IN; k += 4) {
        float2 av = xr2[(k >> 1) + hi];  // A[m][k+2*hi], A[m][k+2*hi+1]
        v2f a;
        a.x = av.x;
        a.y = av.y;
#pragma unroll
        for (int j = 0; j < NT; ++j) {
            const float* wcol = Wm + col0 + j * 16 + m;
            v2f b;
            b.x = wcol[(k + 2 * hi) * ldw];
            b.y = wcol[(k + 2 * hi + 1) * ldw];
            acc[j] = __builtin_amdgcn_wmma_f32_16x16x4_f32(
                false, a, false, b, (short)0, acc[j], false, false);
        }
    }

    // C/D layout: VGPR v holds row (v + 8*hi), col = lane&15
#pragma unroll
    for (int v = 0; v < 8; ++v) {
        int row = rb * 16 + v + 8 * hi;
        float dv = dinv[row];
#pragma unroll
        for (int j = 0; j < NT; ++j) {
            H[(size_t)row * ldw + col0 + j * 16 + m] = acc[j][v] * dv;
        }
    }
}

// ---------------- buffer copy (self-loop init: agg = h') ----------------
__global__ void copy4_kernel(const float4* __restrict__ src, float4* __restrict__ dst, int n4) {
    int i = blockIdx.x * blockDim.x + threadIdx.x;
    if (i < n4) dst[i] = src[i];
}

// ---------------- edge scatter: agg[dst] += h'[src] ----------------
__global__ void scatter_kernel(const float* __restrict__ H, float* __restrict__ Agg,
                               const int* __restrict__ src, const int* __restrict__ dst,
                               int e, int F, int epb) {
    int eIdx = blockIdx.x * epb + threadIdx.x / F;
    int f = threadIdx.x % F;
    if (eIdx >= e) return;
    int s = src[eIdx];
    int t = dst[eIdx];
    unsafeAtomicAdd(&Agg[(size_t)t * F + f], H[(size_t)s * F + f]);
}

// ---------------- fused: y = agg*dinv + bias (in place) + per-feature stats ----
__global__ void bn_stats_kernel(float* __restrict__ Y, const float* __restrict__ dinv,
                                const float* __restrict__ bias, float* __restrict__ stats,
                                int n, int rowsPerBlock) {
    int f = threadIdx.x & 127;
    int half = threadIdx.x >> 7;
    int start = blockIdx.x * rowsPerBlock;
    int end = min(n, start + rowsPerBlock);
    float bf = bias[f];
    float s = 0.0f, ss = 0.0f;
    for (int r = start + half; r < end; r += 2) {
        float y = Y[(size_t)r * 128 + f] * dinv[r] + bf;
        Y[(size_t)r * 128 + f] = y;
        s += y;
        ss += y * y;
    }
    __shared__ float ls[256], lss[256];
    ls[threadIdx.x] = s;
    lss[threadIdx.x] = ss;
    __syncthreads();
    if (half == 0) {
        s += ls[128 + f];
        ss += lss[128 + f];
        unsafeAtomicAdd(&stats[f], s);
        unsafeAtomicAdd(&stats[128 + f], ss);
    }
}

__global__ void zero_stats_kernel(float* stats) { stats[threadIdx.x] = 0.0f; }

__global__ void bn_finalize_kernel(float* stats, const float* __restrict__ g,
                                   const float* __restrict__ be, float invN) {
    int f = threadIdx.x;  // 128 threads
    float mu = stats[f] * invN;
    float var = stats[128 + f] * invN - mu * mu;
    float sc = g[f] * rsqrtf(var + BN_EPS);
    stats[f] = sc;
    stats[128 + f] = be[f] - mu * sc;
}

__global__ void bn_apply_kernel(float* __restrict__ Y, const float* __restrict__ stats, int total) {
    int i = blockIdx.x * blockDim.x + threadIdx.x;
    if (i >= total) return;
    int f = i & 127;
    float y = Y[i] * stats[f] + stats[128 + f];
    Y[i] = fmaxf(y, 0.0f);
}

// ---------------- final: z = agg*dinv + b3 (40 cols), log_softmax ----------------
__global__ void final_kernel(const float* __restrict__ Agg, const float* __restrict__ dinv,
                             const float* __restrict__ b3, float* __restrict__ out, int n) {
    int wave = threadIdx.x >> 5;
    int lane = threadIdx.x & 31;
    int i = blockIdx.x * 8 + wave;
    if (i >= n) return;
    float dv = dinv[i];
    float z0 = (lane < F_OUT) ? Agg[(size_t)i * F_OUT_PAD + lane] * dv + b3[lane] : -INFINITY;
    float z1 = (lane + 32 < F_OUT) ? Agg[(size_t)i * F_OUT_PAD + lane + 32] * dv + b3[lane + 32]
                                   : -INFINITY;
    float m = fmaxf(z0, z1);
#pragma unroll
    for (int off = 16; off > 0; off >>= 1) m = fmaxf(m, __shfl_xor(m, off, 32));
    float e0 = (lane < F_OUT) ? expf(z0 - m) : 0.0f;
    float e1 = (lane + 32 < F_OUT) ? expf(z1 - m) : 0.0f;
    float s = e0 + e1;
#pragma unroll
    for (int off = 16; off > 0; off >>= 1) s += __shfl_xor(s, off, 32);
    float lse = m + logf(s);
    if (lane < F_OUT) out[(size_t)i * F_OUT + lane] = z0 - lse;
    if (lane + 32 < F_OUT) out[(size_t)i * F_OUT + lane + 32] = z1 - lse;
}

extern "C" void kernel_launch(void* const* d_in, const int* in_sizes, int n_in,
                              void* d_out, int out_size, void* d_ws, size_t ws_size,
                              hipStream_t stream) {
    const float* x   = (const float*)d_in[0];
    const int*   src = (const int*)d_in[1];
    const int*   dst = (const int*)d_in[2];
    const float* W1  = (const float*)d_in[3];
    const float* b1  = (const float*)d_in[4];
    const float* W2  = (const float*)d_in[5];
    const float* b2  = (const float*)d_in[6];
    const float* W3  = (const float*)d_in[7];
    const float* b3  = (const float*)d_in[8];
    const float* g1  = (const float*)d_in[9];
    const float* be1 = (const float*)d_in[10];
    const float* g2  = (const float*)d_in[11];
    const float* be2 = (const float*)d_in[12];
    float* out = (float*)d_out;

    // workspace layout (floats, 256B-aligned offsets)
    float* ws    = (float*)d_ws;
    float* dinv  = ws;                    // N
    float* stats = ws + 100352;           // 256
    float* w3p   = ws + 100608;           // 128*48
    float* bufA  = ws + 106752;           // N*128  (h')
    float* bufB  = bufA + (size_t)N_NODES * F_HID;  // N*128 (agg / y / next X)

    const int nRB = N_NODES / 16;  // 6250 row blocks
    dim3 blk256(256);

    // --- dinv ---
    deg_init_kernel<<<(N_NODES + 255) / 256, blk256, 0, stream>>>(dinv, N_NODES);
    deg_accum_kernel<<<(N_EDGES + 255) / 256, blk256, 0, stream>>>(dinv, dst, N_EDGES);
    dinv_kernel<<<(N_NODES + 255) / 256, blk256, 0, stream>>>(dinv, N_NODES);
    pad_w3_kernel<<<(F_HID * F_OUT_PAD + 255) / 256, blk256, 0, stream>>>(W3, w3p);

    const int gemmBlocksX = (nRB + 7) / 8;
    const int statsBlocks = 500, rowsPerBlock = 200;  // 500*200 = 100000

    // ---------- Layer 1: X = x ----------
    gemm_dinv_kernel<4><<<dim3(gemmBlocksX, 2), blk256, 0, stream>>>(x, W1, dinv, bufA, F_HID, nRB);
    copy4_kernel<<<(N_NODES * F_HID / 4 + 255) / 256, blk256, 0, stream>>>(
        (const float4*)bufA, (float4*)bufB, N_NODES * F_HID / 4);
    scatter_kernel<<<N_EDGES / 2, dim3(256), 0, stream>>>(bufA, bufB, src, dst, N_EDGES, 128, 2);
    zero_stats_kernel<<<1, 256, 0, stream>>>(stats);
    bn_stats_kernel<<<statsBlocks, blk256, 0, stream>>>(bufB, dinv, b1, stats, N_NODES, rowsPerBlock);
    bn_finalize_kernel<<<1, 128, 0, stream>>>(stats, g1, be1, 1.0f / N_NODES);
    bn_apply_kernel<<<(N_NODES * F_HID + 255) / 256, blk256, 0, stream>>>(bufB, stats,
                                                                          N_NODES * F_HID);

    // ---------- Layer 2: X = bufB ----------
    gemm_dinv_kernel<4><<<dim3(gemmBlocksX, 2), blk256, 0, stream>>>(bufB, W2, dinv, bufA, F_HID,
                                                                     nRB);
    copy4_kernel<<<(N_NODES * F_HID / 4 + 255) / 256, blk256, 0, stream>>>(
        (const float4*)bufA, (float4*)bufB, N_NODES * F_HID / 4);
    scatter_kernel<<<N_EDGES / 2, dim3(256), 0, stream>>>(bufA, bufB, src, dst, N_EDGES, 128, 2);
    zero_stats_kernel<<<1, 256, 0, stream>>>(stats);
    bn_stats_kernel<<<statsBlocks, blk256, 0, stream>>>(bufB, dinv, b2, stats, N_NODES, rowsPerBlock);
    bn_finalize_kernel<<<1, 128, 0, stream>>>(stats, g2, be2, 1.0f / N_NODES);
    bn_apply_kernel<<<(N_NODES * F_HID + 255) / 256, blk256, 0, stream>>>(bufB, stats,
                                                                          N_NODES * F_HID);

    // ---------- Layer 3: X = bufB, W = w3p (128x48) ----------
    // bufB is read by the GEMM, then overwritten by the copy (self-loop init) afterwards.
    gemm_dinv_kernel<3><<<dim3(gemmBlocksX, 1), blk256, 0, stream>>>(bufB, w3p, dinv, bufA,
                                                                     F_OUT_PAD, nRB);
    copy4_kernel<<<(N_NODES * F_OUT_PAD / 4 + 255) / 256, blk256, 0, stream>>>(
        (const float4*)bufA, (float4*)bufB, N_NODES * F_OUT_PAD / 4);
    scatter_kernel<<<(N_EDGES + 4) / 5, dim3(240), 0, stream>>>(bufA, bufB, src, dst, N_EDGES,
                                                                F_OUT_PAD, 5);
    final_kernel<<<(N_NODES + 7) / 8, blk256, 0, stream>>>(bufB, dinv, b3, out, N_NODES);
}